// SimpleGraphLayer_2714419331079
// MI455X (gfx1250) — compile-verified
//
#include <hip/hip_runtime.h>

#define B_   8
#define C_   128
#define N_   8192
#define K_   16
#define OUT_ 128

typedef __attribute__((ext_vector_type(2))) float v2f;
typedef __attribute__((ext_vector_type(4))) float v4f;
typedef __attribute__((ext_vector_type(8))) float v8f;

// ---------------------------------------------------------------------------
// Kernel 1: transpose x (B,C,N) -> xt (B*N, C) via LDS tiles (coalesced both ways)
// ---------------------------------------------------------------------------
__global__ void tposeBCN_kernel(const float* __restrict__ x, float* __restrict__ xt) {
    __shared__ float tile[32][33];
    const int b     = blockIdx.z;
    const int ctile = blockIdx.y * 32;
    const int ntile = blockIdx.x * 32;
    const int tx = threadIdx.x;   // 0..31 -> n within tile (read), c within tile (write)
    const int ty = threadIdx.y;   // 0..7

    const float* xb = x + (size_t)b * C_ * N_;
#pragma unroll
    for (int i = 0; i < 4; ++i) {
        const int c = ty + i * 8;
        tile[c][tx] = xb[(size_t)(ctile + c) * N_ + (ntile + tx)];
    }
    __syncthreads();
    float* xtb = xt + ((size_t)b * N_ + ntile) * C_ + ctile;
#pragma unroll
    for (int i = 0; i < 4; ++i) {
        const int n = ty + i * 8;
        xtb[(size_t)n * C_ + tx] = tile[tx][n];
    }
}

// ---------------------------------------------------------------------------
// Kernel 2: gather K neighbor rows per point, max-reduce -> agg (B*N, C)
// One wave per point; lane = float4 chunk of the 128-float row (coalesced 512B).
// ---------------------------------------------------------------------------
__global__ void gather_max_kernel(const float* __restrict__ xt,
                                  const int* __restrict__ idx,
                                  float* __restrict__ agg) {
    const int wave = threadIdx.x >> 5;
    const int lane = threadIdx.x & 31;
    const int p    = blockIdx.x * 8 + wave;

    const v4f* __restrict__ xt4 = (const v4f*)xt;
    const int* __restrict__ ip  = idx + (size_t)p * K_;

    v4f acc;
    acc.x = acc.y = acc.z = acc.w = -3.402823466e38f;
#pragma unroll
    for (int k = 0; k < K_; ++k) {
        const int q = ip[k];                      // same addr all lanes -> broadcast
        const v4f v = xt4[(size_t)q * (C_ / 4) + lane];
        acc.x = fmaxf(acc.x, v.x);
        acc.y = fmaxf(acc.y, v.y);
        acc.z = fmaxf(acc.z, v.z);
        acc.w = fmaxf(acc.w, v.w);
    }
    ((v4f*)agg)[(size_t)p * (C_ / 4) + lane] = acc;
}

// ---------------------------------------------------------------------------
// Kernel 3: out[b][o][n] = relu( sum_c W[o][c]*agg[p][c] + bias[o] ), p=b*N+n
// WMMA f32 16x16x4: A = W tile (16 o x 4 c), B = agg^T tile (4 c x 16 p),
// D lanes hold 16 consecutive points -> contiguous stores per output channel.
// One wave per 16x16 tile; 8 waves/block share one point-tile of agg.
// ---------------------------------------------------------------------------
__global__ void gemm_bias_relu_kernel(const float* __restrict__ agg,
                                      const float* __restrict__ W,
                                      const float* __restrict__ bias,
                                      float* __restrict__ out) {
    const int warp   = threadIdx.x >> 5;
    const int lane   = threadIdx.x & 31;
    const int tileId = blockIdx.x * 8 + warp;
    const int obase  = (tileId & 7) * 16;        // 8 output-channel tiles
    const int pbase  = (tileId >> 3) * 16;       // 4096 point tiles

    const int nl = lane & 15;                    // A: row m    B: col n
    const int kh = lane >> 4;                    // which K-pair (k = 2*kh + vgpr)

    const float* __restrict__ wrow = W   + (size_t)(obase + nl) * C_ + 2 * kh;
    const float* __restrict__ arow = agg + (size_t)(pbase + nl) * C_ + 2 * kh;

    v8f acc = {};
#pragma unroll
    for (int cb = 0; cb < C_; cb += 4) {
        const v2f a = *(const v2f*)(wrow + cb);  // A(m, cb+2kh .. +1)
        const v2f bm = *(const v2f*)(arow + cb); // B(cb+2kh .., n) = agg[p][c]
        acc = __builtin_amdgcn_wmma_f32_16x16x4_f32(
            /*neg_a=*/false, a, /*neg_b=*/false, bm,
            /*c_mod=*/(short)0, acc, /*reuse_a=*/false, /*reuse_b=*/false);
    }

    // D layout: vgpr r, lane L -> o = obase + r + 8*(L>>4), p = pbase + (L&15)
    const int p  = pbase + nl;
    const int bq = p >> 13;                      // N_ = 8192 = 2^13
    const int n  = p & (N_ - 1);
    float* __restrict__ outp = out + (size_t)bq * OUT_ * N_ + n;
#pragma unroll
    for (int r = 0; r < 8; ++r) {
        const int o = obase + r + 8 * kh;
        float v = acc[r] + bias[o];
        v = v > 0.0f ? v : 0.0f;
        outp[(size_t)o * N_] = v;
    }
}

// ---------------------------------------------------------------------------
extern "C" void kernel_launch(void* const* d_in, const int* in_sizes, int n_in,
                              void* d_out, int out_size, void* d_ws, size_t ws_size,
                              hipStream_t stream) {
    const float* x    = (const float*)d_in[0];   // (B, C, N) f32
    const int*   idx  = (const int*)d_in[1];     // (B*N*K,) int
    const float* W    = (const float*)d_in[2];   // (OUT, C, 1) f32
    const float* bias = (const float*)d_in[3];   // (OUT,) f32
    float*       out  = (float*)d_out;           // (B, OUT, N) f32

    float* xt  = (float*)d_ws;                   // (B*N, C)  32 MB
    float* agg = xt + (size_t)B_ * N_ * C_;      // (B*N, C)  32 MB

    {
        dim3 tb(32, 8);
        dim3 tg(N_ / 32, C_ / 32, B_);
        tposeBCN_kernel<<<tg, tb, 0, stream>>>(x, xt);
    }
    {
        const int blocks = (B_ * N_) / 8;        // 8 waves (points) per block
        gather_max_kernel<<<blocks, 256, 0, stream>>>(xt, idx, agg);
    }
    {
        const int tiles  = (B_ * N_ / 16) * (OUT_ / 16);  // 32768
        const int blocks = tiles / 8;                      // 8 waves per block
        gemm_bias_relu_kernel<<<blocks, 256, 0, stream>>>(agg, W, bias, out);
    }
}